// LocalSubGraph_68195490726566
// MI455X (gfx1250) — compile-verified
//
#include <hip/hip_runtime.h>

// Fused PointNet-ish LocalSubGraph for gfx1250 (wave32, WMMA fp32).
// B=16, N=128, P=64, D=128, L=3.  One workgroup = 2 polygons (M=128 rows).

#define B_    16
#define N_    128
#define P_    64
#define D_    128
#define L_    3
#define S_    132          // hbuf row stride (floats): breaks A-fragment bank conflicts
#define WS_   288          // wbuf pair-row stride (floats): 288%64==32 -> half-waves on
                           // disjoint bank sets for paired b64 B-fragment loads
#define NT    256          // threads per block = 8 waves
#define NTILE 8            // 16-wide N tiles per wave (8*16 = 128 cols)
#define HROWS 128          // rows held in LDS (2 polygons * 64 points)

typedef __attribute__((ext_vector_type(2))) float v2f;
typedef __attribute__((ext_vector_type(4))) float v4f;
typedef __attribute__((ext_vector_type(8))) float v8f;

// Stage a 128x128 row-major weight block into the WMMA-paired LDS layout:
//   wbuf[(k>>1)*WS_ + 2*c + (k&1)] = W[k][c]
// so that a B fragment {W[k0][c], W[k0+1][c]} is one aligned ds_load_b64.
__device__ __forceinline__ void stage_w_paired(const float* __restrict__ w,
                                               float* __restrict__ wbuf, int t) {
  for (int j = t; j < (D_ * D_ / 4); j += NT) {
    v4f v = *(const v4f*)(w + (size_t)j * 4);
    const int k = j >> 5;            // weight row (K index)
    const int c = (j & 31) * 4;      // column base
    float* dst = wbuf + (k >> 1) * WS_ + (k & 1) + c * 2;
    dst[0] = v.x; dst[2] = v.y; dst[4] = v.z; dst[6] = v.w;
  }
}

__global__ __launch_bounds__(NT)
void psg_fused(const float* __restrict__ x, const unsigned char* __restrict__ msk,
               const float* __restrict__ W1, const float* __restrict__ b1,
               const float* __restrict__ lng, const float* __restrict__ lnb,
               const float* __restrict__ W2, const float* __restrict__ b2,
               float* __restrict__ out)
{
  extern __shared__ float lds[];
  float* hbuf = lds;                    // HROWS * S_      (activations, in-place per layer)
  float* wbuf = hbuf + HROWS * S_;      // 64 * WS_        (current B matrix, paired layout)
  float* aggb = wbuf + 64 * WS_;        // 2 * 128         (masked max per polygon)
  float* rbuf = aggb + 256;             // 2 * 128         (b2 + agg @ W2_bottom)
  unsigned int* mb = (unsigned int*)(rbuf + 256);          // 2 polys * 2 mask words

  const int t    = threadIdx.x;
  const int lane = t & 31;
  const int wave = t >> 5;
  const int colL = lane & 15;           // N index within a 16x16 tile / A-row selector
  const int kh   = lane >> 4;           // half-wave: K-pair / M+8 selector
  const int Mb   = wave * 16;           // this wave's M-tile base row
  const int wpoly= wave >> 2;           // polygon owned by this wave's M-tile
  const long bn0 = (long)blockIdx.x * 2;

  // ---- mask bitmaps + stage x tile (2*64*128 fp32 contiguous) ----
  if (t < 4) mb[t] = 0u;
  __syncthreads();
  if (t < 128) {
    int q = t >> 6, p = t & 63;
    if (msk[(bn0 + q) * P_ + p]) atomicOr(&mb[q*2 + (p >> 5)], 1u << (p & 31));
  }
  {
    const float* xs = x + bn0 * (P_ * D_);
    for (int j = t; j < (HROWS * D_ / 4); j += NT) {
      v4f v = *(const v4f*)(xs + (size_t)j * 4);
      *(v4f*)(hbuf + (j >> 5) * S_ + (j & 31) * 4) = v;
    }
  }
  __syncthreads();

  unsigned int em[2][2]; bool pv[2];
  #pragma unroll
  for (int q = 0; q < 2; q++) {
    unsigned int a = mb[q*2], b = mb[q*2+1];
    pv[q] = (a | b) != 0u;
    em[q][0] = pv[q] ? a : 0xffffffffu;   // fully-invalid poly: treat all valid,
    em[q][1] = pv[q] ? b : 0xffffffffu;   // output forced to 0 at the end
  }

  for (int li = 0; li < L_; ++li) {
    const float* w1  = W1 + (size_t)li * D_ * D_;
    const float* w2t = W2 + (size_t)li * 2 * D_ * D_;   // top half  (h   @ W2_top)
    const float* w2b = w2t + (size_t)D_ * D_;           // bottom    (agg @ W2_bot)

    // ---- stage W1 into LDS (WMMA-paired layout) ----
    stage_w_paired(w1, wbuf, t);
    __syncthreads();

    // ---- GEMM1: t = h @ W1 + b1  (V_WMMA_F32_16X16X4_F32) ----
    v8f acc[NTILE];
    #pragma unroll
    for (int n = 0; n < NTILE; n++) {
      float bv = b1[li * D_ + n * 16 + colL];
      #pragma unroll
      for (int r = 0; r < 8; r++) acc[n][r] = bv;
    }
    {
      const float* arow = hbuf + (Mb + colL) * S_;
      for (int kb = 0; kb < D_; kb += 4) {
        v2f a = *(const v2f*)(arow + kb + 2 * kh);       // A: lane=M, vgpr/half=K
        const float* wp = wbuf + ((kb >> 1) + kh) * WS_ + colL * 2;
        #pragma unroll
        for (int n = 0; n < NTILE; n++) {
          v2f b = *(const v2f*)(wp + n * 32);            // paired b64: rows k0,k0+1
          acc[n] = __builtin_amdgcn_wmma_f32_16x16x4_f32(
              false, a, false, b, (short)0, acc[n], false, false);
        }
      }
    }
    __syncthreads();                                    // all A reads done
    #pragma unroll
    for (int n = 0; n < NTILE; n++)
      #pragma unroll
      for (int r = 0; r < 8; r++)                       // C/D: vgpr=M row, half=+8
        hbuf[(Mb + r + 8 * kh) * S_ + n * 16 + colL] = acc[n][r];
    __syncthreads();

    // ---- LayerNorm + ReLU (threads 0-127) | prefetch next weights (128-255) ----
    if (t < 128) {
      float* row = hbuf + t * S_;
      float s = 0.f;
      for (int e = 0; e < D_; e += 4) {
        v4f v = *(const v4f*)(row + e); s += (v.x + v.y) + (v.z + v.w);
      }
      const float m = s * (1.0f / D_);
      float sv = 0.f;
      for (int e = 0; e < D_; e += 4) {
        v4f v = *(const v4f*)(row + e);
        float a0 = v.x - m, a1 = v.y - m, a2 = v.z - m, a3 = v.w - m;
        sv += (a0*a0 + a1*a1) + (a2*a2 + a3*a3);
      }
      const float is = rsqrtf(sv * (1.0f / D_) + 1e-5f);
      const float* g  = lng + li * D_;
      const float* bb = lnb + li * D_;
      for (int e = 0; e < D_; e += 4) {
        v4f v  = *(const v4f*)(row + e);
        v4f gv = *(const v4f*)(g + e);
        v4f bv = *(const v4f*)(bb + e);
        v.x = fmaxf((v.x - m) * is * gv.x + bv.x, 0.f);
        v.y = fmaxf((v.y - m) * is * gv.y + bv.y, 0.f);
        v.z = fmaxf((v.z - m) * is * gv.z + bv.z, 0.f);
        v.w = fmaxf((v.w - m) * is * gv.w + bv.w, 0.f);
        *(v4f*)(row + e) = v;
      }
    } else if (li + 1 < L_) {
      const float* nw1 = W1 + (size_t)(li + 1) * D_ * D_;
      const float* nw2 = W2 + (size_t)(li + 1) * 2 * D_ * D_;
      for (int j = t - 128; j < 512;  j += 128) __builtin_prefetch(nw1 + j * 32, 0, 1);
      for (int j = t - 128; j < 1024; j += 128) __builtin_prefetch(nw2 + j * 32, 0, 1);
    }
    __syncthreads();

    // ---- masked max over the 64 points of each polygon ----
    {
      const int q = t >> 7, e = t & 127;
      const float* hb = hbuf + q * (P_ * S_) + e;
      float mx = -__builtin_inff();
      for (int p = 0; p < P_; p++)
        if ((em[q][p >> 5] >> (p & 31)) & 1u) mx = fmaxf(mx, hb[p * S_]);
      aggb[t] = mx;
    }
    __syncthreads();
    // ---- r = b2 + agg @ W2_bottom  (1x128 matvec, broadcast over points) ----
    {
      const int q = t >> 7, e = t & 127;
      const float* ag = aggb + q * 128;
      float s = b2[li * D_ + e];
      for (int d = 0; d < D_; d++) s += ag[d] * w2b[(size_t)d * D_ + e];
      rbuf[t] = s;
    }
    // ---- stage W2_top into LDS (W1 no longer needed) ----
    stage_w_paired(w2t, wbuf, t);
    __syncthreads();

    // ---- GEMM2: h = h1 @ W2_top + r ----
    #pragma unroll
    for (int n = 0; n < NTILE; n++) {
      float rv = rbuf[wpoly * 128 + n * 16 + colL];
      #pragma unroll
      for (int r = 0; r < 8; r++) acc[n][r] = rv;
    }
    {
      const float* arow = hbuf + (Mb + colL) * S_;
      for (int kb = 0; kb < D_; kb += 4) {
        v2f a = *(const v2f*)(arow + kb + 2 * kh);
        const float* wp = wbuf + ((kb >> 1) + kh) * WS_ + colL * 2;
        #pragma unroll
        for (int n = 0; n < NTILE; n++) {
          v2f b = *(const v2f*)(wp + n * 32);
          acc[n] = __builtin_amdgcn_wmma_f32_16x16x4_f32(
              false, a, false, b, (short)0, acc[n], false, false);
        }
      }
    }
    __syncthreads();
    #pragma unroll
    for (int n = 0; n < NTILE; n++)
      #pragma unroll
      for (int r = 0; r < 8; r++)
        hbuf[(Mb + r + 8 * kh) * S_ + n * 16 + colL] = acc[n][r];
    __syncthreads();
  }

  // ---- final masked max + invalid-polygon zeroing ----
  {
    const int q = t >> 7, e = t & 127;
    const float* hb = hbuf + q * (P_ * S_) + e;
    float mx = -__builtin_inff();
    for (int p = 0; p < P_; p++)
      if ((em[q][p >> 5] >> (p & 31)) & 1u) mx = fmaxf(mx, hb[p * S_]);
    out[(bn0 + q) * D_ + e] = pv[q] ? mx : 0.f;
  }
}

extern "C" void kernel_launch(void* const* d_in, const int* in_sizes, int n_in,
                              void* d_out, int out_size, void* d_ws, size_t ws_size,
                              hipStream_t stream) {
  (void)in_sizes; (void)n_in; (void)d_ws; (void)ws_size; (void)out_size;
  const float*         x  = (const float*)d_in[0];
  const unsigned char* m  = (const unsigned char*)d_in[1];  // jax bool -> 1 byte
  const float*         W1 = (const float*)d_in[2];
  const float*         b1 = (const float*)d_in[3];
  const float*         lg = (const float*)d_in[4];
  const float*         lb = (const float*)d_in[5];
  const float*         W2 = (const float*)d_in[6];
  const float*         b2 = (const float*)d_in[7];
  float*              o   = (float*)d_out;

  const size_t shmem = (size_t)(HROWS * S_ + 64 * WS_ + 512) * sizeof(float)
                     + 4 * sizeof(unsigned int);           // ~143 KB (2 WGs / WGP)
  (void)hipFuncSetAttribute((const void*)psg_fused,
                            hipFuncAttributeMaxDynamicSharedMemorySize, (int)shmem);
  psg_fused<<<(B_ * N_) / 2, NT, shmem, stream>>>(x, m, W1, b1, lg, lb, W2, b2, o);
}